// GaussianScene_19825569039025
// MI455X (gfx1250) — compile-verified
//
#include <hip/hip_runtime.h>
#include <hip/hip_bf16.h>

#define BLOCK 256

// ---- CDNA5 async global->LDS copies (ASYNCcnt path, bypasses VGPRs) ----
__device__ __forceinline__ void async_ld_b128(unsigned lds_byte_off, const float* gaddr) {
    asm volatile("global_load_async_to_lds_b128 %0, %1, off"
                 :: "v"(lds_byte_off), "v"(gaddr) : "memory");
}
__device__ __forceinline__ void async_ld_b32(unsigned lds_byte_off, const float* gaddr) {
    asm volatile("global_load_async_to_lds_b32 %0, %1, off"
                 :: "v"(lds_byte_off), "v"(gaddr) : "memory");
}
__device__ __forceinline__ void wait_async0() {
    asm volatile("s_wait_asynccnt 0" ::: "memory");
}

// On gfx1250 the shared aperture occupies addr[63:32]; the low 32 bits of a
// generic pointer to LDS are the LDS byte offset (CDNA5 ISA).
__device__ __forceinline__ unsigned lds_off(const void* p) {
    return (unsigned)(uintptr_t)p;
}

__global__ __launch_bounds__(BLOCK) void gs_project_kernel(
    const float* __restrict__ pts,   // (N,3)
    const float* __restrict__ quat,  // (N,4)
    const float* __restrict__ scl,   // (N,3)
    const float* __restrict__ E,     // (4,4) extrinsic
    const float* __restrict__ K,     // (3,4) intrinsic
    float* __restrict__ out_pp,      // (N,2)
    float* __restrict__ out_z,       // (N,1)
    float* __restrict__ out_cov,     // (N,2,2)
    int N)
{
    __shared__ __align__(16) float sP[BLOCK * 3];
    __shared__ __align__(16) float sS[BLOCK * 3];

    const int t    = threadIdx.x;
    const int base = blockIdx.x * BLOCK;
    if (base >= N) return;
    const int nv   = (N - base) < BLOCK ? (N - base) : BLOCK;
    const int nfl  = 3 * nv;
    const int n4   = nfl >> 2;   // whole float4s
    const int rem  = nfl & 3;    // tail floats

    const float* gP = pts + (size_t)base * 3;
    const float* gS = scl + (size_t)base * 3;

    // Stage stride-12B AoS arrays into LDS with async B128 DMA copies.
    if (t < n4) {
        async_ld_b128(lds_off(&sP[4 * t]), gP + 4 * t);
        async_ld_b128(lds_off(&sS[4 * t]), gS + 4 * t);
    }
    if (t < rem) {
        async_ld_b32(lds_off(&sP[4 * n4 + t]), gP + 4 * n4 + t);
        async_ld_b32(lds_off(&sS[4 * n4 + t]), gS + 4 * n4 + t);
    }

    // Uniform camera matrices -> scalar (SMEM) loads while DMA is in flight.
    const float e00 = E[0],  e01 = E[1],  e02 = E[2],  e03 = E[3];
    const float e10 = E[4],  e11 = E[5],  e12 = E[6],  e13 = E[7];
    const float e20 = E[8],  e21 = E[9],  e22 = E[10], e23 = E[11];
    const float e30 = E[12], e31 = E[13], e32 = E[14], e33 = E[15];
    const float k00 = K[0],  k01 = K[1],  k02 = K[2],  k03 = K[3];
    const float k10 = K[4],  k11 = K[5],  k12 = K[6],  k13 = K[7];
    const float k20 = K[8],  k21 = K[9],  k22 = K[10], k23 = K[11];

    wait_async0();
    __syncthreads();

    if (t >= nv) return;
    const int i = base + t;

    // Quaternion: naturally 16B/point -> direct coalesced b128 load.
    const float4 q = reinterpret_cast<const float4*>(quat)[(size_t)i];

    // LDS reads: lane stride of 3 dwords across 64 banks -> conflict free.
    const float px = sP[3 * t + 0], py = sP[3 * t + 1], pz = sP[3 * t + 2];
    const float sx = sS[3 * t + 0], sy = sS[3 * t + 1], sz = sS[3 * t + 2];

    // ---- normalize quaternion ----
    const float inv = 1.0f / sqrtf(q.x * q.x + q.y * q.y + q.z * q.z + q.w * q.w);
    const float w = q.x * inv, x = q.y * inv, y = q.z * inv, z = q.w * inv;

    // ---- rotation matrix ----
    const float r00 = 1.0f - 2.0f * (y * y + z * z);
    const float r01 = 2.0f * (x * y - w * z);
    const float r02 = 2.0f * (x * z + w * y);
    const float r10 = 2.0f * (x * y + w * z);
    const float r11 = 1.0f - 2.0f * (x * x + z * z);
    const float r12 = 2.0f * (y * z - w * x);
    const float r20 = 2.0f * (x * z - w * y);
    const float r21 = 2.0f * (y * z + w * x);
    const float r22 = 1.0f - 2.0f * (x * x + y * y);

    // ---- M = R * diag-broadcast(scales) ; cov3d = M M^T (symmetric) ----
    const float m00 = r00 * sx, m01 = r01 * sy, m02 = r02 * sz;
    const float m10 = r10 * sx, m11 = r11 * sy, m12 = r12 * sz;
    const float m20 = r20 * sx, m21 = r21 * sy, m22 = r22 * sz;

    const float c00 = m00 * m00 + m01 * m01 + m02 * m02;
    const float c01 = m00 * m10 + m01 * m11 + m02 * m12;
    const float c02 = m00 * m20 + m01 * m21 + m02 * m22;
    const float c11 = m10 * m10 + m11 * m11 + m12 * m12;
    const float c12 = m10 * m20 + m11 * m21 + m12 * m22;
    const float c22 = m20 * m20 + m21 * m21 + m22 * m22;

    // ---- camera transform (cam = E @ [p,1]) ----
    const float camX = e00 * px + e01 * py + e02 * pz + e03;
    const float camY = e10 * px + e11 * py + e12 * pz + e13;
    const float camZ = e20 * px + e21 * py + e22 * pz + e23;
    const float camW = e30 * px + e31 * py + e32 * pz + e33;
    const float invW = 1.0f / camW;
    const float X = camX * invW, Y = camY * invW, Z = camZ * invW;

    // ---- projection ----
    const float p0 = k00 * X + k01 * Y + k02 * Z + k03;
    const float p1 = k10 * X + k11 * Y + k12 * Z + k13;
    const float p2 = k20 * X + k21 * Y + k22 * Z + k23;
    const float invP2 = 1.0f / p2;
    const float u = p0 * invP2, v = p1 * invP2;

    // ---- Jacobian J and T = J @ E[:3,:3] (rows 0,1 only; row 2 is zero) ----
    const float fx = k00, fy = k11;
    const float invZ  = 1.0f / Z;
    const float invZ2 = invZ * invZ;
    const float j00 = fx * invZ,  j02 = -fx * X * invZ2;
    const float j11 = fy * invZ,  j12 = -fy * Y * invZ2;

    const float t00 = j00 * e00 + j02 * e20;
    const float t01 = j00 * e01 + j02 * e21;
    const float t02 = j00 * e02 + j02 * e22;
    const float t10 = j11 * e10 + j12 * e20;
    const float t11 = j11 * e11 + j12 * e21;
    const float t12 = j11 * e12 + j12 * e22;

    // ---- projected covariance: T2 @ cov3d @ T2^T ----
    const float v00 = t00 * c00 + t01 * c01 + t02 * c02;
    const float v01 = t00 * c01 + t01 * c11 + t02 * c12;
    const float v02 = t00 * c02 + t01 * c12 + t02 * c22;
    const float v10 = t10 * c00 + t11 * c01 + t12 * c02;
    const float v11 = t10 * c01 + t11 * c11 + t12 * c12;
    const float v12 = t10 * c02 + t11 * c12 + t12 * c22;

    const float P00 = v00 * t00 + v01 * t01 + v02 * t02;
    const float P01 = v00 * t10 + v01 * t11 + v02 * t12;
    const float P10 = v10 * t00 + v11 * t01 + v12 * t02;
    const float P11 = v10 * t10 + v11 * t11 + v12 * t12;

    // ---- coalesced stores ----
    reinterpret_cast<float2*>(out_pp)[(size_t)i] = make_float2(u, v);
    out_z[(size_t)i] = camZ;  // z component pre-divide
    reinterpret_cast<float4*>(out_cov)[(size_t)i] = make_float4(P00, P01, P10, P11);
}

extern "C" void kernel_launch(void* const* d_in, const int* in_sizes, int n_in,
                              void* d_out, int out_size, void* d_ws, size_t ws_size,
                              hipStream_t stream) {
    const float* pts  = (const float*)d_in[0];
    const float* quat = (const float*)d_in[1];
    const float* scl  = (const float*)d_in[2];
    const float* E    = (const float*)d_in[3];
    const float* K    = (const float*)d_in[4];

    const int N = in_sizes[0] / 3;

    float* out_pp  = (float*)d_out;                  // 2N floats
    float* out_z   = out_pp + (size_t)2 * N;         // N floats
    float* out_cov = out_pp + (size_t)3 * N;         // 4N floats

    const int grid = (N + BLOCK - 1) / BLOCK;
    gs_project_kernel<<<grid, BLOCK, 0, stream>>>(pts, quat, scl, E, K,
                                                  out_pp, out_z, out_cov, N);
}